// NCC_87686052315723
// MI455X (gfx1250) — compile-verified
//
#include <hip/hip_runtime.h>

typedef __attribute__((ext_vector_type(2))) float v2f;
typedef __attribute__((ext_vector_type(8))) float v8f;

#define IMG_H   1024
#define IMG_W   1024
#define NIMG    8
#define PADW    4
#define WINW    9
#define TR      16              // output rows per block
#define TC      64              // output cols per block (4 waves x 16 cols)
#define IR      (TR + 2*PADW)   // 24 input rows
#define ICA     80              // aligned staging cols: c0-8 .. c0+71 (float4 aligned)
#define NVEC    (ICA / 4)       // 20 float4 per staged row
#define NTHREADS 128            // 4 wave32 waves
#define GRID_X  (IMG_W / TC)    // 16
#define GRID_Y  (IMG_H / TR)    // 64
#define NBLOCKS (GRID_X * GRID_Y * NIMG)  // 8192

__global__ __launch_bounds__(NTHREADS) void ncc_tile_kernel(
    const float* __restrict__ y_pred,   // Ji
    const float* __restrict__ y_true,   // Ii
    float* __restrict__ partial)
{
  __shared__ float sI[IR][ICA];
  __shared__ float sJ[IR][ICA];
  __shared__ float sH[5][IR][TC];   // horizontal 9-sums: I, J, I2, J2, IJ
  __shared__ float sRed[NTHREADS];

  const int tid = threadIdx.x;
  const int r0  = blockIdx.y * TR;
  const int c0  = blockIdx.x * TC;
  const size_t base = (size_t)blockIdx.z * (size_t)IMG_H * IMG_W;

  // ---- Phase 1: cooperative 128-bit loads into aligned LDS staging ----
  // Staged global cols: gc = c0 - 8 + [0, ICA); 16B-aligned since c0 % 64 == 0.
  for (int idx = tid; idx < IR * NVEC; idx += NTHREADS) {
    const int rr  = idx / NVEC;
    const int v   = idx % NVEC;
    const int gr  = r0 - PADW + rr;
    const int gc0 = c0 - 2 * PADW + 4 * v;
    const bool row_ok = (gr >= 0) && (gr < IMG_H);
    float4 iv = make_float4(0.f, 0.f, 0.f, 0.f);
    float4 jv = make_float4(0.f, 0.f, 0.f, 0.f);
    if (row_ok && gc0 >= 0 && gc0 + 3 < IMG_W) {
      const size_t off = base + (size_t)gr * IMG_W + (size_t)gc0;
      iv = *reinterpret_cast<const float4*>(y_true + off);   // global_load_b128
      jv = *reinterpret_cast<const float4*>(y_pred + off);
    } else if (row_ok) {
      // image-border tiles only: per-component guarded loads
      float ic[4] = {0.f, 0.f, 0.f, 0.f};
      float jc[4] = {0.f, 0.f, 0.f, 0.f};
#pragma unroll
      for (int k = 0; k < 4; ++k) {
        const int gc = gc0 + k;
        if (gc >= 0 && gc < IMG_W) {
          const size_t off = base + (size_t)gr * IMG_W + (size_t)gc;
          ic[k] = y_true[off];
          jc[k] = y_pred[off];
        }
      }
      iv = make_float4(ic[0], ic[1], ic[2], ic[3]);
      jv = make_float4(jc[0], jc[1], jc[2], jc[3]);
    }
    *reinterpret_cast<float4*>(&sI[rr][4 * v]) = iv;   // ds_store_b128
    *reinterpret_cast<float4*>(&sJ[rr][4 * v]) = jv;
  }
  __syncthreads();

  // ---- Phase 2: horizontal 9-tap sliding sums of the 5 quantities ----
  // Output col cc covers global cols c0+cc-4 .. c0+cc+4 -> LDS cols cc+4 .. cc+12.
  for (int idx = tid; idx < IR * TC; idx += NTHREADS) {
    const int rr = idx / TC;
    const int cc = idx % TC;
    float sA = 0.f, sB = 0.f, sA2 = 0.f, sB2 = 0.f, sAB = 0.f;
#pragma unroll
    for (int d = 0; d < WINW; ++d) {
      const float a = sI[rr][cc + PADW + d];
      const float b = sJ[rr][cc + PADW + d];
      sA += a;
      sB += b;
      sA2 = fmaf(a, a, sA2);
      sB2 = fmaf(b, b, sB2);
      sAB = fmaf(a, b, sAB);
    }
    sH[0][rr][cc] = sA;
    sH[1][rr][cc] = sB;
    sH[2][rr][cc] = sA2;
    sH[3][rr][cc] = sB2;
    sH[4][rr][cc] = sAB;
  }
  __syncthreads();

  // ---- Phase 3: vertical 9-sum as band-matrix multiply via WMMA f32 16x16x4 ----
  // Y[16x16] = Wv[16x24] x H[24x16],  Wv[m][k] = 1 iff m <= k <= m+8
  const int wave = tid >> 5;        // wave32
  const int lane = tid & 31;
  const int m    = lane & 15;
  const int hi   = lane >> 4;       // 0: lanes 0-15, 1: lanes 16-31
  const int wc   = wave * 16;       // this wave's 16 output columns in tile

  // A operand (16x4 f32 layout): lanes 0-15 hold K=0(V0),1(V1); lanes 16-31 K=2,3
  v2f a[6];
#pragma unroll
  for (int kc = 0; kc < 6; ++kc) {
    const int k0 = 4 * kc + 2 * hi;
    a[kc].x = (k0     >= m && k0     <= m + 2 * PADW) ? 1.0f : 0.0f;
    a[kc].y = (k0 + 1 >= m && k0 + 1 <= m + 2 * PADW) ? 1.0f : 0.0f;
  }

  v8f acc[5] = {};
#pragma unroll
  for (int kc = 0; kc < 6; ++kc) {
#pragma unroll
    for (int q = 0; q < 5; ++q) {
      // B operand (4x16 f32 layout): V0 holds rows K,K+2 ; V1 holds rows K+1,K+3
      v2f b;
      b.x = sH[q][4 * kc + 2 * hi    ][wc + m];
      b.y = sH[q][4 * kc + 2 * hi + 1][wc + m];
      acc[q] = __builtin_amdgcn_wmma_f32_16x16x4_f32(
          false, a[kc], false, b, (short)0, acc[q], false, false);
    }
  }

  // ---- Phase 4: pointwise NCC math + per-lane partial sum ----
  const float inv_win = 1.0f / (float)(WINW * WINW);
  float lsum = 0.0f;
#pragma unroll
  for (int v = 0; v < 8; ++v) {
    const float Is  = acc[0][v];
    const float Js  = acc[1][v];
    const float I2s = acc[2][v];
    const float J2s = acc[3][v];
    const float IJs = acc[4][v];
    const float uI = Is * inv_win;
    const float uJ = Js * inv_win;
    const float cross = fmaf(-uI, uJ, IJs * inv_win);
    const float Ivar  = fmaf(-uI, uI, I2s * inv_win);
    const float Jvar  = fmaf(-uJ, uJ, J2s * inv_win);
    const float den   = fmaxf(Ivar * Jvar, 1e-7f);
    lsum += cross / __builtin_sqrtf(den);
  }

  // ---- Phase 5: deterministic block reduction -> per-block partial ----
  sRed[tid] = lsum;
  __syncthreads();
  for (int s = NTHREADS / 2; s > 0; s >>= 1) {
    if (tid < s) sRed[tid] += sRed[tid + s];
    __syncthreads();
  }
  if (tid == 0) {
    const int blin = ((int)blockIdx.z * (int)gridDim.y + (int)blockIdx.y) * (int)gridDim.x
                     + (int)blockIdx.x;
    partial[blin] = sRed[0];
  }
}

__global__ __launch_bounds__(256) void ncc_finalize_kernel(
    const float* __restrict__ partial, float* __restrict__ out)
{
  __shared__ float sRed[256];
  const int tid = threadIdx.x;
  float s = 0.0f;
  for (int i = tid; i < NBLOCKS; i += 256) s += partial[i];
  sRed[tid] = s;
  __syncthreads();
  for (int st = 128; st > 0; st >>= 1) {
    if (tid < st) sRed[tid] += sRed[tid + st];
    __syncthreads();
  }
  if (tid == 0) {
    const float mean = sRed[0] / (float)((size_t)NIMG * IMG_H * IMG_W);
    out[0] = 1.0f - mean;
  }
}

extern "C" void kernel_launch(void* const* d_in, const int* in_sizes, int n_in,
                              void* d_out, int out_size, void* d_ws, size_t ws_size,
                              hipStream_t stream) {
  (void)in_sizes; (void)n_in; (void)out_size; (void)ws_size;
  const float* y_pred = (const float*)d_in[0];
  const float* y_true = (const float*)d_in[1];
  float* partial = (float*)d_ws;    // NBLOCKS floats, fully rewritten each call

  dim3 grid(GRID_X, GRID_Y, NIMG);
  ncc_tile_kernel<<<grid, NTHREADS, 0, stream>>>(y_pred, y_true, partial);
  ncc_finalize_kernel<<<1, 256, 0, stream>>>(partial, (float*)d_out);
}